// EncoderChannels_27273042330242
// MI455X (gfx1250) — compile-verified
//
#include <hip/hip_runtime.h>
#include <hip/hip_bf16.h>

typedef float v2f __attribute__((ext_vector_type(2)));
typedef float v8f __attribute__((ext_vector_type(8)));

// D = A(16x4 f32) * B(4x16 f32) + C, full f32 precision on the matrix pipe.
__device__ __forceinline__ v8f wmma_f32(v2f a, v2f b, v8f c) {
  return __builtin_amdgcn_wmma_f32_16x16x4_f32(false, a, false, b, (short)0, c,
                                               false, false);
}

// ---------------------------------------------------------------------------
// Problem constants
// ---------------------------------------------------------------------------
#define BATCH   8
#define CIN     3
#define HW      320
#define CLS     2
#define PSZ     10
#define P2      100     // PSZ*PSZ
#define LPATCH  1024    // (320/10)^2
#define HID     200     // 2*P2

// Workspace layout (floats)
#define WS_INI  0
#define WS_FEA  (WS_INI + BATCH*CLS*HW*HW)            // 1,638,400 each
#define WS_UI   (WS_FEA + BATCH*CLS*HW*HW)
#define WS_UF   (WS_UI  + BATCH*CLS*LPATCH*P2)
#define WS_W1P  (WS_UF  + BATCH*CLS*LPATCH*P2)        // [208][100] zero-padded
#define WS_W2P  (WS_W1P + 208*100)                    // [112][200] zero-padded

// ---------------------------------------------------------------------------
// Kernel 0: zero-pad the MLP weights so trans GEMM K-loops are guard-free.
// ---------------------------------------------------------------------------
__global__ __launch_bounds__(256) void pad_weights_kernel(
    const float* __restrict__ w_lin1, const float* __restrict__ w_lin2,
    float* __restrict__ w1p, float* __restrict__ w2p) {
  int i = blockIdx.x * 256 + threadIdx.x;
  if (i < 208 * 100) w1p[i] = (i < HID * P2) ? w_lin1[i] : 0.f;
  if (i < 112 * 200) w2p[i] = (i < P2 * HID) ? w_lin2[i] : 0.f;
}

// ---------------------------------------------------------------------------
// Kernel 1: fused conv1(7x7,3->64) -> conv2(3x3,64->2) + fm(1x1)/4
// One block per 16x16 output tile. conv1 is an implicit GEMM on WMMA:
//   M = 324 halo pixels (18x18, padded to 336), N = 64, K = 147 (padded 148).
// A-gather uses a precomputed k->offset LDS table; off(147) points at a
// dedicated zero region appended to the x tile, so the K tail needs no guard.
// LDS floats: xs 2160 | offtab 148 | w1s 9472 | b1s 64 | t1s 64*325
//             | w2s 1152 | cs 16
// ---------------------------------------------------------------------------
#define XS_FLOATS        2160           // 1728 data + 432 zeros
#define CONV_LDS_FLOATS  (XS_FLOATS + 148 + 9472 + 64 + 64*325 + 1152 + 16)

__global__ __launch_bounds__(256) void conv_fused_kernel(
    const float* __restrict__ x,  const float* __restrict__ w1,
    const float* __restrict__ b1, const float* __restrict__ w2,
    const float* __restrict__ b2, const float* __restrict__ wfm,
    const float* __restrict__ bfm,
    float* __restrict__ ini, float* __restrict__ fea) {
  extern __shared__ float smem[];
  float* xs     = smem;                    // [3][24][24] + 432-float zero pad
  int*   offtab = (int*)(xs + XS_FLOATS);  // [148] A-gather offsets (8B align)
  float* w1s    = xs + XS_FLOATS + 148;    // [74][64][2] K-pair interleaved B
  float* b1s    = w1s + 74 * 64 * 2;       // [64]
  float* t1s    = b1s + 64;                // [64][325] conv1 (stride 325, odd)
  float* w2s    = t1s + 64 * 325;          // [2][64][9]
  float* cs     = w2s + 1152;              // b2[2], wfm[6], bfm[2]

  const int tid = threadIdx.x;
  const int X0 = blockIdx.x * 16, Y0 = blockIdx.y * 16, b = blockIdx.z;

  // ---- stage inputs into LDS --------------------------------------------
  for (int i = tid; i < XS_FLOATS; i += 256) {
    float v = 0.f;
    if (i < CIN * 576) {
      int c = i / 576, r = i % 576, iy = r / 24, ix = r % 24;
      int gy = Y0 - 4 + iy, gx = X0 - 4 + ix;
      if (gy >= 0 && gy < HW && gx >= 0 && gx < HW)
        v = x[((size_t)(b * CIN + c) * HW + gy) * HW + gx];
    }
    xs[i] = v;
  }
  if (tid < 148) {
    int off = 1728;                         // zero region
    if (tid < 147) {
      int c = tid / 49, rr = tid % 49;
      off = c * 576 + (rr / 7) * 24 + (rr % 7);
    }
    offtab[tid] = off;
  }
  for (int i = tid; i < 148 * 64; i += 256) {
    int k = i >> 6, n = i & 63;
    // pair-interleaved: element ((k/2)*64 + n)*2 + (k&1)
    w1s[(((k >> 1) << 6) + n) * 2 + (k & 1)] =
        (k < 147) ? w1[n * 147 + k] : 0.f;  // B[k][n] = w1[n][k]
  }
  if (tid < 64) b1s[tid] = b1[tid];
  for (int i = tid; i < CLS * 64 * 9; i += 256) w2s[i] = w2[i];
  if (tid < 2) cs[tid]     = b2[tid];
  if (tid < 6) cs[2 + tid] = wfm[tid];
  if (tid < 2) cs[8 + tid] = bfm[tid];
  __syncthreads();

  // ---- conv1 as WMMA implicit GEMM --------------------------------------
  const int wave  = tid >> 5;
  const int lane  = tid & 31;
  const int l16   = lane & 15;
  const int lhalf = (lane < 16) ? 0 : 2;     // K offset of this lane half

  for (int t = wave; t < 21 * 4; t += 8) {   // 21 M-tiles x 4 N-tiles
    const int m0 = (t >> 2) * 16, n0 = (t & 3) * 16;
    const int m  = m0 + l16;
    const int mm = (m < 324) ? m : 0;        // invalid rows masked at store
    const int abase = (mm / 18) * 24 + (mm % 18);
    const int2* ot = ((const int2*)offtab) + (lhalf >> 1);
    const v2f*  bp = ((const v2f*)w1s) + (lhalf >> 1) * 64 + n0 + l16;
    v8f acc = {};
    #pragma unroll 4
    for (int k0 = 0; k0 < 148; k0 += 4) {
      int2 o = *ot;  ot += 2;                // {off[ka], off[ka+1]}
      v2f a;
      a.x = xs[abase + o.x];
      a.y = xs[abase + o.y];
      v2f bf = *bp;  bp += 128;              // {B[ka][n], B[ka+1][n]}
      acc = wmma_f32(a, bf, acc);
    }
    // D: VGPR r -> (M = m0+r (+8 upper half), N = n0 + l16)
    const int ch = n0 + l16;
    const float bias = b1s[ch];
    #pragma unroll
    for (int r = 0; r < 8; ++r) {
      int M = m0 + r + ((lane < 16) ? 0 : 8);
      if (M < 324) {
        int ty = M / 18, tx = M % 18;
        int gy = Y0 - 1 + ty, gx = X0 - 1 + tx;
        // conv2 zero-pads: conv1 values outside the image must read as 0
        float v = (gy >= 0 && gy < HW && gx >= 0 && gx < HW) ? (acc[r] + bias)
                                                             : 0.f;
        t1s[ch * 325 + M] = v;
      }
    }
  }
  __syncthreads();

  // ---- conv2 (N=2: VALU) + fm/4 -----------------------------------------
  {
    const int oy = tid >> 4, ox = tid & 15;   // 256 threads == 16x16 tile
    float a0 = cs[0], a1 = cs[1];
    for (int ci = 0; ci < 64; ++ci) {
      const float* trow = &t1s[ci * 325];
      const float* wr0  = &w2s[ci * 9];
      const float* wr1  = &w2s[(64 + ci) * 9];
      #pragma unroll
      for (int dy = 0; dy < 3; ++dy)
        #pragma unroll
        for (int dx = 0; dx < 3; ++dx) {
          float tv = trow[(oy + dy) * 18 + (ox + dx)];
          a0 = fmaf(wr0[dy * 3 + dx], tv, a0);
          a1 = fmaf(wr1[dy * 3 + dx], tv, a1);
        }
    }
    const int gy = Y0 + oy, gx = X0 + ox;
    const size_t o0 = ((size_t)(b * CLS + 0) * HW + gy) * HW + gx;
    const size_t o1 = ((size_t)(b * CLS + 1) * HW + gy) * HW + gx;
    ini[o0] = a0;
    ini[o1] = a1;
    float x0 = xs[(oy + 4) * 24 + (ox + 4)];
    float x1 = xs[576 + (oy + 4) * 24 + (ox + 4)];
    float x2 = xs[1152 + (oy + 4) * 24 + (ox + 4)];
    fea[o0] = (cs[2] * x0 + cs[3] * x1 + cs[4] * x2 + cs[8]) * 0.25f;
    fea[o1] = (cs[5] * x0 + cs[6] * x1 + cs[7] * x2 + cs[9]) * 0.25f;
  }
}

// ---------------------------------------------------------------------------
// Kernel 2: unfold + per-patch MLP (Linear 100->200, Linear 200->100, ReLU)
// 64 patches per block; GEMM1: M=64,K=100,N=208pad; GEMM2: M=64,K=200,N=112pad.
// Padded weights make every K-step guard-free; even LDS strides give aligned
// ds_load_b64 fragments that stay bank-conflict free.
// LDS floats: a 64*102 | h 64*210
// ---------------------------------------------------------------------------
#define TRANS_LDS_FLOATS (64 * 102 + 64 * 210)

__global__ __launch_bounds__(256) void trans_kernel(
    const float* __restrict__ ini, const float* __restrict__ fea,
    const float* __restrict__ w1p, const float* __restrict__ w2p,
    float* __restrict__ ui, float* __restrict__ uf) {
  extern __shared__ float smem[];
  float* a_lds = smem;               // [64][102]
  float* h_lds = a_lds + 64 * 102;   // [64][210] (cols 0..207 written)

  const int tid = threadIdx.x;
  const float* src = blockIdx.y ? fea : ini;
  float* dst       = blockIdx.y ? uf  : ui;
  const int P0 = blockIdx.x * 64;    // first patch row of this block

  // unfold 64 patches directly from [B,C,H,W]
  for (int i = tid; i < 64 * P2; i += 256) {
    int r = i / P2, p = i % P2;
    int Pa = P0 + r;
    int bc = Pa >> 10, l = Pa & 1023;
    int gy = (l >> 5) * PSZ + p / PSZ;
    int gx = (l & 31) * PSZ + p % PSZ;
    a_lds[r * 102 + p] = src[((size_t)bc * HW + gy) * HW + gx];
  }
  __syncthreads();

  const int wave  = tid >> 5;
  const int lane  = tid & 31;
  const int l16   = lane & 15;
  const int lhalf = (lane < 16) ? 0 : 2;

  // GEMM1: h = t * W1^T   (4 M-tiles x 13 N-tiles; padded rows give acc==0)
  for (int t = wave; t < 4 * 13; t += 8) {
    const int m0 = (t / 13) * 16, n0 = (t % 13) * 16;
    const int n  = n0 + l16;
    const v2f* ap = (const v2f*)&a_lds[(m0 + l16) * 102 + lhalf];
    const v2f* bp = (const v2f*)&w1p[n * P2 + lhalf];
    v8f acc = {};
    #pragma unroll 5
    for (int k0 = 0; k0 < P2; k0 += 4) {
      v2f a  = *ap;  ap += 2;
      v2f bf = *bp;  bp += 2;
      acc = wmma_f32(a, bf, acc);
    }
    #pragma unroll
    for (int r = 0; r < 8; ++r) {
      int M = m0 + r + ((lane < 16) ? 0 : 8);
      h_lds[M * 210 + n] = acc[r];           // cols >=200 are exact zeros
    }
  }
  __syncthreads();

  // GEMM2: out = relu(h * W2^T)   (4 M-tiles x 7 N-tiles)
  for (int t = wave; t < 4 * 7; t += 8) {
    const int m0 = (t / 7) * 16, n0 = (t % 7) * 16;
    const int n  = n0 + l16;
    const v2f* ap = (const v2f*)&h_lds[(m0 + l16) * 210 + lhalf];
    const v2f* bp = (const v2f*)&w2p[n * HID + lhalf];
    v8f acc = {};
    #pragma unroll 5
    for (int k0 = 0; k0 < HID; k0 += 4) {
      v2f a  = *ap;  ap += 2;
      v2f bf = *bp;  bp += 2;
      acc = wmma_f32(a, bf, acc);
    }
    if (n < P2) {
      #pragma unroll
      for (int r = 0; r < 8; ++r) {
        int M = m0 + r + ((lane < 16) ? 0 : 8);
        float v = acc[r];
        dst[(size_t)(P0 + M) * P2 + n] = v > 0.f ? v : 0.f;
      }
    }
  }
}

// ---------------------------------------------------------------------------
// Kernel 3: att = ui * uf^T / 100 per (b,c). 128x64 output tile per block.
// LDS floats: U 128*102 | V 64*102
// ---------------------------------------------------------------------------
#define ATT_LDS_FLOATS (128 * 102 + 64 * 102)

__global__ __launch_bounds__(256) void att_kernel(
    const float* __restrict__ ui, const float* __restrict__ uf,
    float* __restrict__ out) {
  extern __shared__ float smem[];
  float* us = smem;              // [128][102]
  float* vs = us + 128 * 102;    // [64][102]

  const int tid = threadIdx.x;
  const int bc = blockIdx.z;
  const int M0 = blockIdx.x * 128, N0 = blockIdx.y * 64;
  const float* U = ui + (size_t)bc * LPATCH * P2;
  const float* V = uf + (size_t)bc * LPATCH * P2;

  for (int i = tid; i < 128 * P2; i += 256) {
    int r = i / P2, k = i % P2;
    us[r * 102 + k] = U[(size_t)(M0 + r) * P2 + k];
  }
  for (int i = tid; i < 64 * P2; i += 256) {
    int r = i / P2, k = i % P2;
    vs[r * 102 + k] = V[(size_t)(N0 + r) * P2 + k];
  }
  __syncthreads();

  const int wave  = tid >> 5;
  const int lane  = tid & 31;
  const int l16   = lane & 15;
  const int lhalf = (lane < 16) ? 0 : 2;

  for (int t = wave; t < 8 * 4; t += 8) {   // 8 M-tiles x 4 N-tiles
    const int m0 = (t >> 2) * 16, n0 = (t & 3) * 16;
    const v2f* ap = (const v2f*)&us[(m0 + l16) * 102 + lhalf];
    const v2f* bp = (const v2f*)&vs[(n0 + l16) * 102 + lhalf];  // B[k][n]=V[n][k]
    v8f acc = {};
    #pragma unroll 5
    for (int k0 = 0; k0 < P2; k0 += 4) {
      v2f a  = *ap;  ap += 2;
      v2f bf = *bp;  bp += 2;
      acc = wmma_f32(a, bf, acc);
    }
    #pragma unroll
    for (int r = 0; r < 8; ++r) {
      int M = M0 + m0 + r + ((lane < 16) ? 0 : 8);
      int N = N0 + n0 + l16;
      out[((size_t)bc * LPATCH + M) * LPATCH + N] = acc[r] * 0.01f;
    }
  }
}

// ---------------------------------------------------------------------------
extern "C" void kernel_launch(void* const* d_in, const int* in_sizes, int n_in,
                              void* d_out, int out_size, void* d_ws,
                              size_t ws_size, hipStream_t stream) {
  (void)in_sizes; (void)n_in; (void)out_size; (void)ws_size;
  const float* x       = (const float*)d_in[0];
  const float* w_conv1 = (const float*)d_in[1];
  const float* b_conv1 = (const float*)d_in[2];
  const float* w_conv2 = (const float*)d_in[3];
  const float* b_conv2 = (const float*)d_in[4];
  const float* w_fm    = (const float*)d_in[5];
  const float* b_fm    = (const float*)d_in[6];
  const float* w_lin1  = (const float*)d_in[7];
  const float* w_lin2  = (const float*)d_in[8];
  float* out = (float*)d_out;

  float* ws  = (float*)d_ws;
  float* ini = ws + WS_INI;
  float* fea = ws + WS_FEA;
  float* ui  = ws + WS_UI;
  float* uf  = ws + WS_UF;
  float* w1p = ws + WS_W1P;
  float* w2p = ws + WS_W2P;

  pad_weights_kernel<<<dim3((112 * 200 + 255) / 256), 256, 0, stream>>>(
      w_lin1, w_lin2, w1p, w2p);

  conv_fused_kernel<<<dim3(HW / 16, HW / 16, BATCH), 256,
                      CONV_LDS_FLOATS * sizeof(float), stream>>>(
      x, w_conv1, b_conv1, w_conv2, b_conv2, w_fm, b_fm, ini, fea);

  trans_kernel<<<dim3(BATCH * CLS * LPATCH / 64, 2, 1), 256,
                 TRANS_LDS_FLOATS * sizeof(float), stream>>>(
      ini, fea, w1p, w2p, ui, uf);

  att_kernel<<<dim3(LPATCH / 128, LPATCH / 64, BATCH * CLS), 256,
               ATT_LDS_FLOATS * sizeof(float), stream>>>(ui, uf, out);
}